// HeteroGNN_64673617543713
// MI455X (gfx1250) — compile-verified
//
#include <hip/hip_runtime.h>
#include <hip/hip_bf16.h>
#include <math.h>

typedef __attribute__((ext_vector_type(2))) float v2f;
typedef __attribute__((ext_vector_type(8))) float v8f;

// ---------------------------------------------------------------------------
// router features: x_router[i,:] = emb[i % 9, :]
__global__ void tile_router_k(float* __restrict__ out, const float* __restrict__ emb, int n) {
  int t = blockIdx.x * blockDim.x + threadIdx.x;
  if (t >= n * 32) return;
  int i = t >> 5, j = t & 31;
  out[t] = emb[(i % 9) * 32 + j];
}

// y[i, j] = b0[j] (+ b1[j]) (+ b2[j])  -- sums the per-relation biases for a dst type
__global__ void bias_init_k(float* __restrict__ y, long long tot, int O,
                            const float* __restrict__ b0,
                            const float* __restrict__ b1,
                            const float* __restrict__ b2) {
  long long t = (long long)blockIdx.x * blockDim.x + threadIdx.x;
  if (t >= tot) return;
  int j = (int)(t % O);
  float v = b0[j];
  if (b1) v += b1[j];
  if (b2) v += b2[j];
  y[t] = v;
}

// C[n,O] (+)= A[n,K] @ W[O,K]^T  using V_WMMA_F32_16X16X4_F32.
// W staged in LDS once per block (8 waves, 128 rows). K-loop manually unrolled x2
// with ping-pong A/B fragment buffers: no rotation moves, no tail branch
// (prefetch index wraps with (ks+8)&(K-1); K is a power of two).
template<bool ACC, int O, int K>
__global__ void wmma_gemm_k(float* __restrict__ C, const float* __restrict__ A,
                            const float* __restrict__ W, int n) {
  constexpr int NTILE = O / 16;
  __shared__ float sW[O * K];
  const int tid = threadIdx.x;

  // cooperative W -> LDS copy (row-major, contiguous)
  #pragma unroll 1
  for (int idx = tid * 4; idx < O * K; idx += 256 * 4)
    *(float4*)(sW + idx) = *(const float4*)(W + idx);
  __syncthreads();

  const int wave = tid >> 5;                  // 8 waves per block
  const int lane = tid & 31;
  const int row0 = (blockIdx.x * 8 + wave) * 16;
  if (row0 >= n) return;                      // wave-uniform exit (EXEC all-ones in WMMA)
  const int half = lane >> 4;                 // 0: lanes 0-15, 1: lanes 16-31
  const int l16  = lane & 15;

  v8f c[NTILE];
  if (ACC) {
    #pragma unroll
    for (int t = 0; t < NTILE; ++t) {
      const float* cp = C + (size_t)(row0 + half * 8) * O + t * 16 + l16;
      #pragma unroll
      for (int i = 0; i < 8; ++i) c[t][i] = cp[(size_t)i * O];
    }
  } else {
    #pragma unroll
    for (int t = 0; t < NTILE; ++t) {
      #pragma unroll
      for (int i = 0; i < 8; ++i) c[t][i] = 0.0f;
    }
  }

  // A 16x4 fragment: lane<16 holds A[row+l16][ks..ks+1], lane>=16 holds [ks+2..ks+3]
  // B 4x16 fragment: B[kk][nn] = W[col0+nn][ks+kk]
  const float* aP = A + (size_t)(row0 + l16) * K + half * 2;
  const float* bP = sW + l16 * K + half * 2;

  auto loadA = [&](v2f& a, int ks) { a[0] = aP[ks]; a[1] = aP[ks + 1]; };
  auto loadB = [&](v2f* b, int ks) {
    #pragma unroll
    for (int t = 0; t < NTILE; ++t) {
      b[t][0] = bP[t * 16 * K + ks];
      b[t][1] = bP[t * 16 * K + ks + 1];
    }
  };
  auto mm = [&](const v2f& a, const v2f* b) {
    #pragma unroll
    for (int t = 0; t < NTILE; ++t)
      c[t] = __builtin_amdgcn_wmma_f32_16x16x4_f32(false, a, false, b[t],
                                                   (short)0, c[t], false, false);
  };

  v2f aX, aY, bX[NTILE], bY[NTILE];
  loadA(aX, 0); loadB(bX, 0);
  #pragma unroll 1
  for (int ks = 0; ks < K; ks += 8) {         // K % 8 == 0 (K = 32 or 128)
    loadA(aY, ks + 4); loadB(bY, ks + 4);     // always in range
    mm(aX, bX);
    const int nxt = (ks + 8) & (K - 1);       // wraps to 0 on last iter (harmless re-read)
    loadA(aX, nxt); loadB(bX, nxt);
    mm(aY, bY);
  }

  #pragma unroll
  for (int t = 0; t < NTILE; ++t) {
    float* cp = C + (size_t)(row0 + half * 8) * O + t * 16 + l16;
    #pragma unroll
    for (int i = 0; i < 8; ++i) cp[(size_t)i * O] = c[t][i];
  }
}

// tiny-O (final layer, O==2) GEMM: one thread per row
__global__ void gemm_small_k(float* __restrict__ C, const float* __restrict__ A,
                             const float* __restrict__ W, int n, int K, int O, int acc) {
  int i = blockIdx.x * blockDim.x + threadIdx.x;
  if (i >= n) return;
  const float* a = A + (size_t)i * K;
  for (int j = 0; j < O; ++j) {
    float s = acc ? C[(size_t)i * O + j] : 0.0f;
    const float* w = W + (size_t)j * K;
    for (int k = 0; k < K; ++k) s += a[k] * w[k];
    C[(size_t)i * O + j] = s;
  }
}

// y[dst[e], j..j+3] += m[src[e], j..j+3]   (O multiple of 4)
__global__ void scatter_add4_k(float* __restrict__ y, const float* __restrict__ m,
                               const int* __restrict__ src, const int* __restrict__ dst,
                               long long tot, int O) {
  long long t = (long long)blockIdx.x * blockDim.x + threadIdx.x;
  if (t >= tot) return;
  int per = O >> 2;
  int e = (int)(t / per), j = (int)(t % per) << 2;
  const float* mp = m + (size_t)src[e] * O + j;
  float* yp = y + (size_t)dst[e] * O + j;
  float4 v = *(const float4*)mp;
  atomicAdd(yp + 0, v.x);
  atomicAdd(yp + 1, v.y);
  atomicAdd(yp + 2, v.z);
  atomicAdd(yp + 3, v.w);
}

// scalar fallback (O==2)
__global__ void scatter_add_k(float* __restrict__ y, const float* __restrict__ m,
                              const int* __restrict__ src, const int* __restrict__ dst,
                              long long tot, int O) {
  long long t = (long long)blockIdx.x * blockDim.x + threadIdx.x;
  if (t >= tot) return;
  int e = (int)(t / O), j = (int)(t % O);
  atomicAdd(&y[(size_t)dst[e] * O + j], m[(size_t)src[e] * O + j]);
}

__global__ void fill_u32_k(unsigned* __restrict__ p, long long n, unsigned v) {
  long long t = (long long)blockIdx.x * blockDim.x + threadIdx.x;
  if (t < n) p[t] = v;
}

__device__ __forceinline__ unsigned f2o(float f) {  // order-preserving float->uint
  unsigned u = __float_as_uint(f);
  return (u & 0x80000000u) ? ~u : (u | 0x80000000u);
}

// agg[dst[e], j..j+3] = max(agg, x[src[e], j..j+3])  encoded as ordered uints (K mult of 4)
__global__ void scatter_max4_k(unsigned* __restrict__ agg, const float* __restrict__ x,
                               const int* __restrict__ src, const int* __restrict__ dst,
                               long long tot, int K) {
  long long t = (long long)blockIdx.x * blockDim.x + threadIdx.x;
  if (t >= tot) return;
  int per = K >> 2;
  int e = (int)(t / per), j = (int)(t % per) << 2;
  const float* xp = x + (size_t)src[e] * K + j;
  unsigned* ap = agg + (size_t)dst[e] * K + j;
  float4 v = *(const float4*)xp;
  atomicMax(ap + 0, f2o(v.x));
  atomicMax(ap + 1, f2o(v.y));
  atomicMax(ap + 2, f2o(v.z));
  atomicMax(ap + 3, f2o(v.w));
}

// decode ordered-uint back to float; untouched (-inf) segments -> 0
__global__ void decode_max_k(float* __restrict__ a, long long n) {
  long long t = (long long)blockIdx.x * blockDim.x + threadIdx.x;
  if (t >= n) return;
  unsigned u = __float_as_uint(a[t]);
  float f = (u & 0x80000000u) ? __uint_as_float(u ^ 0x80000000u) : __uint_as_float(~u);
  if (u == 0x007FFFFFu) f = 0.0f;   // encoding of -inf (empty segment)
  a[t] = f;
}

__global__ void relu_k(float* __restrict__ x, long long n) {
  long long t = (long long)blockIdx.x * blockDim.x + threadIdx.x;
  if (t < n) x[t] = fmaxf(x[t], 0.0f);
}

// ---------------------------------------------------------------------------
static inline unsigned blks(long long n, int t) { return (unsigned)((n + t - 1) / t); }

static inline void launch_gemm(float* C, const float* A, const float* W,
                               int n, int K, int O, bool acc, hipStream_t s) {
  if (O == 128) {
    dim3 grid(blks(n, 128)), block(256);   // 8 waves x 16 rows per block
    if (K == 128) {
      if (acc) wmma_gemm_k<true, 128, 128><<<grid, block, 0, s>>>(C, A, W, n);
      else     wmma_gemm_k<false, 128, 128><<<grid, block, 0, s>>>(C, A, W, n);
    } else {   // K == 32
      if (acc) wmma_gemm_k<true, 128, 32><<<grid, block, 0, s>>>(C, A, W, n);
      else     wmma_gemm_k<false, 128, 32><<<grid, block, 0, s>>>(C, A, W, n);
    }
  } else {
    gemm_small_k<<<blks(n, 256), 256, 0, s>>>(C, A, W, n, K, O, acc ? 1 : 0);
  }
}

extern "C" void kernel_launch(void* const* d_in, const int* in_sizes, int n_in,
                              void* d_out, int out_size, void* d_ws, size_t ws_size,
                              hipStream_t stream) {
  // relation tables: 0 depends_on(max) 1 rev_depends_on 2 mapped_to 3 rev_mapped_to
  //                  4 link 5 interface 6 rev_interface     types: 0 task 1 pe 2 router
  const int RST[7] = {0, 0, 0, 1, 2, 2, 1};
  const int RDT[7] = {0, 0, 1, 0, 2, 1, 2};

  const int NTn = in_sizes[0] / 32;   // 131072 tasks
  const int NPn = in_sizes[1] / 32;   // 36864 PEs
  const int NRn = NPn;                // 36864 routers (batch*9)
  const int NN[3] = {NTn, NPn, NRn};

  const float* x_task = (const float*)d_in[0];
  const float* x_pe   = (const float*)d_in[1];
  const float* r_emb  = (const float*)d_in[2];

  auto Wrel  = [&](int l, int r) { return (const float*)d_in[3 + (l * 7 + r) * 3 + 0]; };
  auto brel  = [&](int l, int r) { return (const float*)d_in[3 + (l * 7 + r) * 3 + 1]; };
  auto Wroot = [&](int l, int r) { return (const float*)d_in[3 + (l * 7 + r) * 3 + 2]; };

  int EC[7]; const int* ES[7]; const int* ED[7];
  for (int r = 0; r < 7; ++r) {
    EC[r] = in_sizes[66 + r] / 2;          // edges[name] is [2, E]: row0=src, row1=dst
    ES[r] = (const int*)d_in[66 + r];
    ED[r] = ES[r] + EC[r];
  }

  // workspace layout (floats)
  float* ws = (float*)d_ws;
  size_t off = 0;
  float* bufT[2]; float* bufP[2]; float* bufR[2];
  bufT[0] = ws + off; off += (size_t)NTn * 128;
  bufT[1] = ws + off; off += (size_t)NTn * 128;
  bufP[0] = ws + off; off += (size_t)NPn * 128;
  bufP[1] = ws + off; off += (size_t)NPn * 128;
  bufR[0] = ws + off; off += (size_t)NRn * 128;
  bufR[1] = ws + off; off += (size_t)NRn * 128;
  float* scratch  = ws + off; off += (size_t)NTn * 128;  // messages / max-agg
  float* router32 = ws + off; off += (size_t)NRn * 32;

  tile_router_k<<<blks((long long)NRn * 32, 256), 256, 0, stream>>>(router32, r_emb, NRn);

  float* outSeg[3] = { (float*)d_out,
                       (float*)d_out + (size_t)NTn * 2,
                       (float*)d_out + (size_t)NTn * 2 + (size_t)NPn * 2 };

  const float* x[3] = { x_task, x_pe, router32 };

  for (int l = 0; l < 3; ++l) {
    const int K = (l == 0) ? 32 : 128;
    const int O = (l == 2) ? 2 : 128;
    float* y[3];
    if (l == 2) { y[0] = outSeg[0]; y[1] = outSeg[1]; y[2] = outSeg[2]; }
    else        { y[0] = bufT[l];   y[1] = bufP[l];   y[2] = bufR[l];   }

    // init y with summed per-relation biases per dst type
    bias_init_k<<<blks((long long)NTn * O, 256), 256, 0, stream>>>(
        y[0], (long long)NTn * O, O, brel(l, 0), brel(l, 1), brel(l, 3));
    bias_init_k<<<blks((long long)NPn * O, 256), 256, 0, stream>>>(
        y[1], (long long)NPn * O, O, brel(l, 2), brel(l, 5), nullptr);
    bias_init_k<<<blks((long long)NRn * O, 256), 256, 0, stream>>>(
        y[2], (long long)NRn * O, O, brel(l, 4), brel(l, 6), nullptr);

    // root contributions: y[dt] += x[dt] @ W_root^T  (one per relation)
    for (int r = 0; r < 7; ++r) {
      int dt = RDT[r];
      launch_gemm(y[dt], x[dt], Wroot(l, r), NN[dt], K, O, /*acc=*/true, stream);
    }

    // message paths
    for (int r = 0; r < 7; ++r) {
      int st = RST[r], dt = RDT[r];
      if (r == 0) {
        // depends_on: segment_max in source dims, then transform
        long long aggTot = (long long)NN[dt] * K;
        fill_u32_k<<<blks(aggTot, 256), 256, 0, stream>>>((unsigned*)scratch, aggTot, 0x007FFFFFu);
        long long tot = (long long)EC[r] * (K >> 2);
        scatter_max4_k<<<blks(tot, 256), 256, 0, stream>>>((unsigned*)scratch, x[st],
                                                           ES[r], ED[r], tot, K);
        decode_max_k<<<blks(aggTot, 256), 256, 0, stream>>>(scratch, aggTot);
        launch_gemm(y[dt], scratch, Wrel(l, r), NN[dt], K, O, /*acc=*/true, stream);
      } else {
        // sum reduce: transform per source node, then scatter-add in O dims
        launch_gemm(scratch, x[st], Wrel(l, r), NN[st], K, O, /*acc=*/false, stream);
        if ((O & 3) == 0) {
          long long tot = (long long)EC[r] * (O >> 2);
          scatter_add4_k<<<blks(tot, 256), 256, 0, stream>>>(y[dt], scratch,
                                                             ES[r], ED[r], tot, O);
        } else {
          long long tot = (long long)EC[r] * O;
          scatter_add_k<<<blks(tot, 256), 256, 0, stream>>>(y[dt], scratch,
                                                            ES[r], ED[r], tot, O);
        }
      }
    }

    if (l < 2) {
      for (int t = 0; t < 3; ++t) {
        long long tot = (long long)NN[t] * O;
        relu_k<<<blks(tot, 256), 256, 0, stream>>>(y[t], tot);
      }
      x[0] = y[0]; x[1] = y[1]; x[2] = y[2];
    }
  }
}